// CACMemoryBank_49649821942413
// MI455X (gfx1250) — compile-verified
//
#include <hip/hip_runtime.h>
#include <hip/hip_bf16.h>
#include <stdint.h>

// Problem constants (from reference shapes)
#define N_Q    1024
#define D_DIM  512
#define BANKN  65536
#define TOTALC (N_Q + BANKN)     // 66560 columns
#define KSEL   3328              // max(1, int(66560 * 0.05))
#define NBINS  512
#define KCHUNK 128
#define NCHUNK (D_DIM / KCHUNK)  // 4
#define AK_STRIDE 132            // 128 + 4 pad dwords: conflict-free ds_load_b64
#define ABUF_DW (16 * AK_STRIDE) // 2112 dwords per A buffer (incl. trailing pad)

typedef __attribute__((ext_vector_type(2))) float v2f;
typedef __attribute__((ext_vector_type(8))) float v8f;
typedef __attribute__((ext_vector_type(4))) unsigned int v4u;
typedef __attribute__((ext_vector_type(4))) int v4i;
typedef __attribute__((ext_vector_type(8))) int v8i;

// ---------------------------------------------------------------------------
// Reciprocal L2 norms for all 66560 rows (queries then bank).
// ---------------------------------------------------------------------------
__global__ __launch_bounds__(256) void cac_rnorm_kernel(
    const float* __restrict__ inputs, const float* __restrict__ bank,
    float* __restrict__ rnorm) {
  const int r = blockIdx.x;
  const float* src = (r < N_Q) ? (inputs + (size_t)r * D_DIM)
                               : (bank + (size_t)(r - N_Q) * D_DIM);
  float s = 0.f;
  for (int k = threadIdx.x; k < D_DIM; k += 256) {
    float v = src[k];
    s += v * v;
  }
  __shared__ float red[256];
  red[threadIdx.x] = s;
  __syncthreads();
  for (int off = 128; off > 0; off >>= 1) {
    if (threadIdx.x < off) red[threadIdx.x] += red[threadIdx.x + off];
    __syncthreads();
  }
  if (threadIdx.x == 0) rnorm[r] = 1.0f / fmaxf(sqrtf(red[0]), 1e-12f);
}

// ---------------------------------------------------------------------------
// TDM issue: async-DMA one 16-row x KCHUNK-col fp32 A chunk into LDS.
// D# pad fields insert 4 dwords after every 128 dwords -> LDS stride 132,
// matching the bank-conflict-free layout the WMMA A reads use.
// ---------------------------------------------------------------------------
__device__ __forceinline__ void tdm_load_a_chunk(uint64_t gaddr,
                                                 uint32_t lds_addr) {
  v4u g0;
  g0[0] = 1u;                                   // count=1, user descriptor
  g0[1] = lds_addr;                             // LDS byte offset (wg-relative)
  g0[2] = (uint32_t)gaddr;                      // global_addr[31:0]
  g0[3] = (uint32_t)(gaddr >> 32) | (2u << 30); // global_addr[56:32] | type=2

  v8i g1;
  g1[0] = (int)((2u << 16)     // data_size = 4 bytes
              | (1u << 20)     // pad_enable
              | (6u << 22)     // pad_interval: every 128 dwords
              | (3u << 25));   // pad_amount: 4 dwords
  g1[1] = (int)(((uint32_t)D_DIM & 0xffffu) << 16);        // tensor_dim0 lo16
  g1[2] = (int)(((uint32_t)D_DIM >> 16) |
                (((uint32_t)N_Q & 0xffffu) << 16));        // dim0 hi | dim1 lo
  g1[3] = (int)(((uint32_t)N_Q >> 16) |
                ((uint32_t)KCHUNK << 16));                 // dim1 hi | tile_dim0
  g1[4] = 16;                                              // tile_dim1=16 rows
  g1[5] = D_DIM;                                           // dim0_stride lo32
  g1[6] = 0;
  g1[7] = 0;

  v4i z4 = {0, 0, 0, 0};
#if defined(__clang_major__) && __clang_major__ >= 23
  v8i z8 = {0, 0, 0, 0, 0, 0, 0, 0};
  __builtin_amdgcn_tensor_load_to_lds(g0, g1, z4, z4, z8, 0);
#else
  __builtin_amdgcn_tensor_load_to_lds(g0, g1, z4, z4, 0);
#endif
}

// ---------------------------------------------------------------------------
// WMMA similarity kernel, two passes:
//   PASS 1: per-row histogram of similarities (selection pre-pass)
//   PASS 2: count label matches above / at the per-row threshold bin
// Block = 256 threads (8 waves): 16 rows x 512 cols. Each wave: 16x64 strip
// (4 accumulator tiles, A reused x4). A chunks double-buffered via TDM.
// ---------------------------------------------------------------------------
template <int PASS>
__global__ __launch_bounds__(256) void cac_sim_kernel(
    const float* __restrict__ inputs, const float* __restrict__ bank,
    const int* __restrict__ gt, const int* __restrict__ bank_labels,
    const float* __restrict__ rnorm,
    unsigned int* __restrict__ hist,      // pass 1 output
    const int* __restrict__ thr_bin,     // pass 2 input
    unsigned int* __restrict__ m_above,  // pass 2 output
    unsigned int* __restrict__ m_in) {   // pass 2 output
  __shared__ float Alds[2][ABUF_DW];
  __shared__ float ra_sh[16];
  __shared__ unsigned int hist_sh[(PASS == 1) ? 16 * NBINS : 1];
  __shared__ unsigned int above_sh[16];
  __shared__ unsigned int in_sh[16];
  __shared__ int thr_sh[16];
  __shared__ int gt_sh[16];

  const int tid = threadIdx.x;
  const int row_base = blockIdx.y * 16;
  const int col_block = blockIdx.x * 512;

  if (tid < 16) {
    ra_sh[tid] = rnorm[row_base + tid];
    if (PASS == 2) {
      thr_sh[tid] = thr_bin[row_base + tid];
      gt_sh[tid]  = gt[row_base + tid];
      above_sh[tid] = 0u;
      in_sh[tid]    = 0u;
    }
  }
  if (PASS == 1) {
    for (int i = tid; i < 16 * NBINS; i += 256) hist_sh[i] = 0u;
  }

  const int wave = tid >> 5;
  const int lane = tid & 31;
  const int half = lane >> 4;      // which K-pair this lane holds
  const int ln   = lane & 15;      // M index for A operand, N index for B
  const int wave_col = col_block + wave * 64;

  // Per-tile column row-pointers (tiles never straddle query/bank boundary:
  // N_Q is a multiple of the 16-wide tile).
  const float* bptr[4];
  float rb[4];
  int   cg[4];
  int   lab[4];
#pragma unroll
  for (int t = 0; t < 4; ++t) {
    const int c = wave_col + t * 16 + ln;
    cg[t] = c;
    const float* rowp = (c < N_Q) ? (inputs + (size_t)c * D_DIM)
                                  : (bank + (size_t)(c - N_Q) * D_DIM);
    bptr[t] = rowp + 2 * half;
    rb[t] = rnorm[c];
    lab[t] = (PASS == 2) ? ((c < N_Q) ? gt[c] : bank_labels[c - N_Q]) : 0;
  }

  const uint64_t a_gbase =
      (uint64_t)(uintptr_t)(inputs + (size_t)row_base * D_DIM);
  const uint32_t lds_b0 = (uint32_t)(uintptr_t)&Alds[0][0];
  const uint32_t lds_b1 = (uint32_t)(uintptr_t)&Alds[1][0];

  v8f acc0 = {}, acc1 = {}, acc2 = {}, acc3 = {};

  // Prime the pipeline: chunk 0 -> buffer 0 (one TDM issue per workgroup).
  if (wave == 0) {
    tdm_load_a_chunk(a_gbase, lds_b0);
  }

  for (int ci = 0; ci < NCHUNK; ++ci) {
    // Guarantee the buffer being refilled is no longer read by any wave.
    __syncthreads();
    if (wave == 0) {
      if (ci + 1 < NCHUNK) {
        tdm_load_a_chunk(a_gbase + (uint64_t)(ci + 1) * KCHUNK * sizeof(float),
                         ((ci + 1) & 1) ? lds_b1 : lds_b0);
        __builtin_amdgcn_s_wait_tensorcnt(1);  // chunk ci done, ci+1 in flight
      } else {
        __builtin_amdgcn_s_wait_tensorcnt(0);  // last chunk done
      }
    }
    __syncthreads();  // chunk ci visible to all waves

    const int kc = ci * KCHUNK;
    const float* aptr = &Alds[ci & 1][ln * AK_STRIDE + 2 * half];

    // Prefetch next K-chunk of the four B feature rows this lane streams.
    if (ci + 1 < NCHUNK) {
#pragma unroll
      for (int t = 0; t < 4; ++t)
        __builtin_prefetch(bptr[t] + kc + KCHUNK, 0, 0);
    }

#pragma unroll 8
    for (int kk = 0; kk < KCHUNK; kk += 4) {
      const v2f a  = *(const v2f*)(aptr + kk);
      const v2f b0 = *(const v2f*)(bptr[0] + kc + kk);
      const v2f b1 = *(const v2f*)(bptr[1] + kc + kk);
      const v2f b2 = *(const v2f*)(bptr[2] + kc + kk);
      const v2f b3 = *(const v2f*)(bptr[3] + kc + kk);
      acc0 = __builtin_amdgcn_wmma_f32_16x16x4_f32(false, a, false, b0,
                                                   (short)0, acc0, false, false);
      acc1 = __builtin_amdgcn_wmma_f32_16x16x4_f32(false, a, false, b1,
                                                   (short)0, acc1, false, false);
      acc2 = __builtin_amdgcn_wmma_f32_16x16x4_f32(false, a, false, b2,
                                                   (short)0, acc2, false, false);
      acc3 = __builtin_amdgcn_wmma_f32_16x16x4_f32(false, a, false, b3,
                                                   (short)0, acc3, false, false);
    }
  }

  // Process accumulators: C/D layout — VGPR g, lanes 0-15: M=g; lanes 16-31:
  // M=g+8; N = lane&15.
  auto process = [&](const v8f& acc, int t) {
    const float rbv = rb[t];
    const int   cgv = cg[t];
#pragma unroll
    for (int g = 0; g < 8; ++g) {
      const int rl = g + 8 * half;
      const int r  = row_base + rl;
      float sim = acc[g] * ra_sh[rl] * rbv;
      if (cgv == r) sim = -2.0f;  // masked diagonal: dist = +inf
      int bin = (int)((sim + 1.0f) * (NBINS * 0.5f));
      bin = max(0, min(NBINS - 1, bin));
      if (PASS == 1) {
        atomicAdd(&hist_sh[rl * NBINS + bin], 1u);
      } else {
        if (lab[t] == gt_sh[rl]) {
          const int tb = thr_sh[rl];
          if (bin > tb)       atomicAdd(&above_sh[rl], 1u);
          else if (bin == tb) atomicAdd(&in_sh[rl], 1u);
        }
      }
    }
  };
  process(acc0, 0); process(acc1, 1); process(acc2, 2); process(acc3, 3);

  __syncthreads();
  if (PASS == 1) {
    for (int i = tid; i < 16 * NBINS; i += 256) {
      const unsigned int v = hist_sh[i];
      if (v) atomicAdd(&hist[(size_t)(row_base + i / NBINS) * NBINS +
                             (i % NBINS)], v);
    }
  } else {
    if (tid < 16) {
      if (above_sh[tid]) atomicAdd(&m_above[row_base + tid], above_sh[tid]);
      if (in_sh[tid])    atomicAdd(&m_in[row_base + tid], in_sh[tid]);
    }
  }
}

// ---------------------------------------------------------------------------
// Per-row threshold: scan histogram from the top; find bin holding the k-th
// largest similarity, how many ties are needed from it, and its population.
// ---------------------------------------------------------------------------
__global__ __launch_bounds__(256) void cac_threshold_kernel(
    const unsigned int* __restrict__ hist, int* __restrict__ thr_bin,
    int* __restrict__ thr_rem, int* __restrict__ thr_cnt) {
  const int r = blockIdx.x * 256 + threadIdx.x;
  if (r >= N_Q) return;
  const unsigned int* h = hist + (size_t)r * NBINS;
  unsigned int cum = 0, rem = KSEL, cnt = 1;
  int tb = 0;
  for (int b = NBINS - 1; b >= 0; --b) {
    const unsigned int c = h[b];
    if (cum + c >= (unsigned int)KSEL) {
      tb = b; rem = KSEL - cum; cnt = (c ? c : 1u);
      break;
    }
    cum += c;
  }
  thr_bin[r] = tb;
  thr_rem[r] = (int)rem;
  thr_cnt[r] = (int)cnt;
}

// ---------------------------------------------------------------------------
// Final scalar: 1 - mean over rows of (matched_above + matched_in*rem/cnt)/k
// ---------------------------------------------------------------------------
__global__ __launch_bounds__(256) void cac_finalize_kernel(
    const unsigned int* __restrict__ m_above,
    const unsigned int* __restrict__ m_in,
    const int* __restrict__ thr_rem, const int* __restrict__ thr_cnt,
    float* __restrict__ out) {
  __shared__ float red[256];
  float s = 0.f;
  for (int r = threadIdx.x; r < N_Q; r += 256) {
    const float frac = ((float)m_above[r] +
                        (float)m_in[r] * ((float)thr_rem[r] / (float)thr_cnt[r]))
                       / (float)KSEL;
    s += frac;
  }
  red[threadIdx.x] = s;
  __syncthreads();
  for (int off = 128; off > 0; off >>= 1) {
    if (threadIdx.x < off) red[threadIdx.x] += red[threadIdx.x + off];
    __syncthreads();
  }
  if (threadIdx.x == 0) out[0] = 1.0f - red[0] / (float)N_Q;
}

// ---------------------------------------------------------------------------
extern "C" void kernel_launch(void* const* d_in, const int* in_sizes, int n_in,
                              void* d_out, int out_size, void* d_ws,
                              size_t ws_size, hipStream_t stream) {
  (void)in_sizes; (void)n_in; (void)out_size; (void)ws_size;
  const float* inputs      = (const float*)d_in[0];
  const int*   gt          = (const int*)d_in[1];
  const float* bank        = (const float*)d_in[2];
  const int*   bank_labels = (const int*)d_in[3];
  float*       out         = (float*)d_out;

  char* ws = (char*)d_ws;
  size_t o = 0;
  auto alloc = [&](size_t bytes) {
    void* p = ws + o;
    o += (bytes + 255) & ~(size_t)255;
    return p;
  };
  float*        rnorm   = (float*)alloc((size_t)TOTALC * sizeof(float));
  unsigned int* hist    = (unsigned int*)alloc((size_t)N_Q * NBINS * sizeof(unsigned int));
  int*          thr_bin = (int*)alloc(N_Q * sizeof(int));
  int*          thr_rem = (int*)alloc(N_Q * sizeof(int));
  int*          thr_cnt = (int*)alloc(N_Q * sizeof(int));
  unsigned int* m_above = (unsigned int*)alloc(N_Q * sizeof(unsigned int));
  unsigned int* m_in    = (unsigned int*)alloc(N_Q * sizeof(unsigned int));

  // Zero accumulation buffers (ws is poisoned, not re-zeroed between replays).
  hipMemsetAsync(hist, 0, (size_t)N_Q * NBINS * sizeof(unsigned int), stream);
  hipMemsetAsync(m_above, 0, N_Q * sizeof(unsigned int), stream);
  hipMemsetAsync(m_in, 0, N_Q * sizeof(unsigned int), stream);

  cac_rnorm_kernel<<<TOTALC, 256, 0, stream>>>(inputs, bank, rnorm);

  dim3 grid(TOTALC / 512, N_Q / 16);  // 130 x 64 blocks
  cac_sim_kernel<1><<<grid, 256, 0, stream>>>(inputs, bank, gt, bank_labels,
                                              rnorm, hist, nullptr, nullptr,
                                              nullptr);
  cac_threshold_kernel<<<(N_Q + 255) / 256, 256, 0, stream>>>(hist, thr_bin,
                                                              thr_rem, thr_cnt);
  cac_sim_kernel<2><<<grid, 256, 0, stream>>>(inputs, bank, gt, bank_labels,
                                              rnorm, hist, thr_bin, m_above,
                                              m_in);
  cac_finalize_kernel<<<1, 256, 0, stream>>>(m_above, m_in, thr_rem, thr_cnt,
                                             out);
}